// GNNEncoder_20358144983223
// MI455X (gfx1250) — compile-verified
//
#include <hip/hip_runtime.h>

#define N_NODES 100000
#define N_EDGES 1600000
#define N_GRAPHS 64
#define D_IN 128
#define D_HID 256
#define D_OUT 128

typedef __attribute__((ext_vector_type(2))) float v2f;
typedef __attribute__((ext_vector_type(8))) float v8f;

// ---------------------------------------------------------------- async copy helpers
// GLOBAL_LOAD_ASYNC_TO_LDS_B128: VDST = LDS byte offset (VGPR), VADDR = 64-bit global addr.
// Tracked by ASYNCcnt; completion enforced with s_wait_asynccnt.
__device__ __forceinline__ void async_b128_to_lds(void* lds_ptr, const void* gptr) {
    unsigned lds_off = (unsigned)(uintptr_t)lds_ptr;   // flat LDS addr low 32 = group offset
    unsigned long long ga = (unsigned long long)(uintptr_t)gptr;
    asm volatile("global_load_async_to_lds_b128 %0, %1, off"
                 :: "v"(lds_off), "v"(ga) : "memory");
}
__device__ __forceinline__ void wait_asynccnt0() {
    asm volatile("s_wait_asynccnt 0x0" ::: "memory");
}

// ---------------------------------------------------------------- init / degree
__global__ void k_init(float* __restrict__ deg, float* __restrict__ out) {
    int i = blockIdx.x * blockDim.x + threadIdx.x;
    if (i < N_NODES) deg[i] = 1.0f;                 // self-loop contributes 1
    if (i < N_GRAPHS * D_OUT) out[i] = 0.0f;        // zero pooled output
}

__global__ void k_degree(const int* __restrict__ dst, float* __restrict__ deg) {
    int e = blockIdx.x * blockDim.x + threadIdx.x;
    if (e < N_EDGES) atomicAdd(&deg[dst[e]], 1.0f);
}

__global__ void k_dinv(float* __restrict__ deg) {
    int i = blockIdx.x * blockDim.x + threadIdx.x;
    if (i < N_NODES) {
        float d = deg[i];
        deg[i] = (d > 0.0f) ? rsqrtf(d) : 0.0f;
    }
}

// ---------------------------------------------------------------- weight packing
// Bp[(k/2)*N + col] = {B[k][col], B[k+1][col]}  -> B-fragment becomes one b64 load.
template<int K>
__global__ void k_packB(const float* __restrict__ B, float2* __restrict__ Bp, int N) {
    int i = blockIdx.x * blockDim.x + threadIdx.x;
    if (i < (K / 2) * N) {
        int kp = i / N, col = i % N;
        Bp[i] = make_float2(B[(size_t)(2 * kp) * N + col],
                            B[(size_t)(2 * kp + 1) * N + col]);
    }
}

// ---------------------------------------------------------------- fp32 WMMA GEMM
// C[M x N] = A[M x K] * Bp (packed).  One block = 16 rows x N cols, 4 waves.
// Wave w covers cols [w*NT*16, (w+1)*NT*16);  N == 4*NT*16;  M % 16 == 0.
// A-fragment (16x4 f32): lane half 0 holds K={kk,kk+1}, half 1 holds K={kk+2,kk+3}.
// B-fragment (4x16):     VGPR0 = rows {kk, kk+2}, VGPR1 = rows {kk+1, kk+3}.
// C/D (16x16 f32):       VGPR r = row r (lanes 0-15) / row r+8 (lanes 16-31).
template<int K, int NT>
__global__ __launch_bounds__(128) void k_gemm(const float* __restrict__ A,
                                              const float2* __restrict__ Bp,
                                              float* __restrict__ C, int N) {
    // +4 float pad: row stride is a multiple of 16B (b128-aligned) and bank stride 4
    __shared__ __align__(16) float As[16 * (K + 4)];
    const int row0 = blockIdx.x * 16;
    const int tid  = threadIdx.x;

    // async-stage the 16 x K A-tile into LDS, 16B per op, no VGPR round trip
    constexpr int CH = K / 4;                       // b128 chunks per row
    for (int i = tid; i < 16 * CH; i += 128) {
        int r = i / CH, c = i % CH;
        async_b128_to_lds(&As[r * (K + 4) + c * 4],
                          &A[(size_t)(row0 + r) * K + c * 4]);
    }
    wait_asynccnt0();
    __syncthreads();

    const int wave = tid >> 5;
    const int lane = tid & 31;
    const int half = lane >> 4;     // which 16-lane half
    const int l    = lane & 15;

    v8f acc[NT];
    const v8f vzero = {0.f, 0.f, 0.f, 0.f, 0.f, 0.f, 0.f, 0.f};
    #pragma unroll
    for (int t = 0; t < NT; ++t) acc[t] = vzero;

    const int colbase = wave * (NT * 16);
    for (int kk = 0; kk < K; kk += 4) {
        v2f a;
        a.x = As[l * (K + 4) + kk + 2 * half + 0];
        a.y = As[l * (K + 4) + kk + 2 * half + 1];
        #pragma unroll
        for (int t = 0; t < NT; ++t) {
            const int col = colbase + t * 16 + l;
            float2 bv = Bp[(size_t)(kk / 2 + half) * N + col];
            v2f b; b.x = bv.x; b.y = bv.y;
            acc[t] = __builtin_amdgcn_wmma_f32_16x16x4_f32(
                /*neg_a=*/false, a, /*neg_b=*/false, b,
                /*c_mod=*/(short)0, acc[t], /*reuse_a=*/false, /*reuse_b=*/false);
        }
    }

    #pragma unroll
    for (int t = 0; t < NT; ++t) {
        const int col = colbase + t * 16 + l;
        #pragma unroll
        for (int r = 0; r < 8; ++r)
            C[(size_t)(row0 + r + 8 * half) * N + col] = acc[t][r];
    }
}

// ---------------------------------------------------------------- aggregation
// agg[v] = h[v] * dinv[v]^2   (self-loop message, also zero-initializes agg)
template<int D>
__global__ void k_selfinit(const float* __restrict__ h, const float* __restrict__ dinv,
                           float* __restrict__ agg) {
    size_t stride = (size_t)gridDim.x * blockDim.x;
    size_t total  = (size_t)N_NODES * D;
    for (size_t i = (size_t)blockIdx.x * blockDim.x + threadIdx.x; i < total; i += stride) {
        int v = (int)(i / D);
        float di = dinv[v];
        agg[i] = h[i] * di * di;
    }
}

// agg[dst] += h[src] * dinv[src]*dinv[dst]  -- one wave (32 lanes) per edge.
template<int D>
__global__ void k_scatter(const float* __restrict__ h, const float* __restrict__ dinv,
                          const int* __restrict__ src, const int* __restrict__ dst,
                          float* __restrict__ agg) {
    int lane = threadIdx.x & 31;
    int e = blockIdx.x * (blockDim.x >> 5) + (threadIdx.x >> 5);
    if (e >= N_EDGES) return;
    int s = src[e], d = dst[e];
    float norm = dinv[s] * dinv[d];
    const float* hs = h + (size_t)s * D;
    float* ad = agg + (size_t)d * D;
    #pragma unroll
    for (int j = 0; j < D / 32; ++j)
        atomicAdd(&ad[lane + 32 * j], hs[lane + 32 * j] * norm);
}

// x = relu(x + b)   (in place)
template<int D>
__global__ void k_bias_relu(float* __restrict__ x, const float* __restrict__ b) {
    size_t stride = (size_t)gridDim.x * blockDim.x;
    size_t total  = (size_t)N_NODES * D;
    for (size_t i = (size_t)blockIdx.x * blockDim.x + threadIdx.x; i < total; i += stride) {
        float v = x[i] + b[i % D];
        x[i] = v > 0.0f ? v : 0.0f;
    }
}

// ---------------------------------------------------------------- global_add_pool
// batch is sorted: register-accumulate runs of equal graph id, flush with one atomic.
__global__ __launch_bounds__(D_OUT) void k_pool(const float* __restrict__ agg2,
                                                const float* __restrict__ b2,
                                                const int* __restrict__ batch,
                                                float* __restrict__ out) {
    const int f = threadIdx.x;                      // 0..127
    const int v0 = blockIdx.x * 256;
    const int v1 = (v0 + 256 < N_NODES) ? (v0 + 256) : N_NODES;
    if (v0 >= N_NODES) return;
    float bf = b2[f];
    float acc = 0.0f;
    int cur = batch[v0];
    for (int v = v0; v < v1; ++v) {
        int g = batch[v];
        if (g != cur) {
            atomicAdd(&out[cur * D_OUT + f], acc);
            acc = 0.0f;
            cur = g;
        }
        acc += agg2[(size_t)v * D_OUT + f] + bf;
    }
    atomicAdd(&out[cur * D_OUT + f], acc);
}

// ---------------------------------------------------------------- launcher
extern "C" void kernel_launch(void* const* d_in, const int* in_sizes, int n_in,
                              void* d_out, int out_size, void* d_ws, size_t ws_size,
                              hipStream_t stream) {
    const float* x    = (const float*)d_in[0];   // (100000, 128)
    const int*   ei   = (const int*)d_in[1];     // (2, 1600000)
    const int*   bat  = (const int*)d_in[2];     // (100000,)
    const float* W1   = (const float*)d_in[3];   // (128, 256)
    const float* b1   = (const float*)d_in[4];   // (256,)
    const float* W2   = (const float*)d_in[5];   // (256, 128)
    const float* b2   = (const float*)d_in[6];   // (128,)
    const int* srcp = ei;
    const int* dstp = ei + N_EDGES;
    float* out = (float*)d_out;

    // Workspace layout (floats): dinv | bufA (1e5*256) | bufB (1e5*256) | Bp1 | Bp2
    float* ws   = (float*)d_ws;
    float* dinv = ws;
    float* bufA = ws + N_NODES;
    float* bufB = bufA + (size_t)N_NODES * D_HID;
    float2* Bp1 = (float2*)(bufB + (size_t)N_NODES * D_HID);          // (D_IN/2)*D_HID pairs
    float2* Bp2 = Bp1 + (size_t)(D_IN / 2) * D_HID;                   // (D_HID/2)*D_OUT pairs
    float* h1   = bufA;                              // layer-1 dense output
    float* agg1 = bufB;                              // layer-1 aggregation
    float* x2   = bufB;                              // relu(agg1+b1), in place
    float* h2   = bufA;                              // reuses dead h1 space
    float* agg2 = bufA + (size_t)N_NODES * D_OUT;    // second half of bufA

    k_init  <<<(N_NODES + 255) / 256, 256, 0, stream>>>(dinv, out);
    k_degree<<<N_EDGES / 256, 256, 0, stream>>>(dstp, dinv);
    k_dinv  <<<(N_NODES + 255) / 256, 256, 0, stream>>>(dinv);

    // pack weights into K-pair-interleaved layout (one b64 per B-fragment)
    k_packB<D_IN ><<<((D_IN / 2) * D_HID + 255) / 256, 256, 0, stream>>>(W1, Bp1, D_HID);
    k_packB<D_HID><<<((D_HID / 2) * D_OUT + 255) / 256, 256, 0, stream>>>(W2, Bp2, D_OUT);

    // layer 1: h1 = x @ W1 ; agg ; relu(+b1)
    k_gemm<D_IN, 4><<<N_NODES / 16, 128, 0, stream>>>(x, Bp1, h1, D_HID);
    k_selfinit<D_HID><<<8192, 256, 0, stream>>>(h1, dinv, agg1);
    k_scatter<D_HID><<<N_EDGES / 8, 256, 0, stream>>>(h1, dinv, srcp, dstp, agg1);
    k_bias_relu<D_HID><<<8192, 256, 0, stream>>>(x2, b1);

    // layer 2: h2 = x2 @ W2 ; agg
    k_gemm<D_HID, 2><<<N_NODES / 16, 128, 0, stream>>>(x2, Bp2, h2, D_OUT);
    k_selfinit<D_OUT><<<8192, 256, 0, stream>>>(h2, dinv, agg2);
    k_scatter<D_OUT><<<N_EDGES / 8, 256, 0, stream>>>(h2, dinv, srcp, dstp, agg2);

    // pooling (adds b2 per node)
    k_pool<<<(N_NODES + 255) / 256, D_OUT, 0, stream>>>(agg2, b2, bat, out);
}